// PNN_46832323396441
// MI455X (gfx1250) — compile-verified
//
#include <hip/hip_runtime.h>
#include <hip/hip_bf16.h>
#include <math.h>

// ---------------------------------------------------------------------------
// PNN forward, fused, fp32 end-to-end, WMMA (V_WMMA_F32_16X16X4_F32) for all
// three matmuls. One wave (32 lanes) owns a 16-row batch tile.
// Weights are pre-swizzled into per-lane WMMA-B layout (zero-padded) in d_ws
// by a prologue kernel, then staged into LDS once per block, so the GEMM
// inner loops are branch-free: ds_load_b64 + ds_load_b64 + v_wmma.
// ---------------------------------------------------------------------------

typedef __attribute__((ext_vector_type(2))) float v2f;
typedef __attribute__((ext_vector_type(8))) float v8f;

#define N_EMB   26
#define N_DENSE 13
#define NFEAT   39          // 26 + 13
#define EMBED   8
#define K_EMB   208         // 26 * 8
#define DPROD   10
#define HIDDEN  64
#define H_DIM   33          // 2*DPROD + N_DENSE
#define H_PAD   36          // K padded to multiple of 4 (cols 33..35 zero)
#define TILE    16
#define WAVES_PER_BLOCK 2

// packed-weight layout in d_ws / LDS (units: floats)
#define BZ_STEPS  (K_EMB / 4)            // 52
#define LP_STEPS  ((EMBED * EMBED) / 4)  // 16
#define W1_STEPS  (H_PAD / 4)            // 9
#define BZ_OFF    0
#define BZ_FLOATS (BZ_STEPS * 32 * 2)            // 3328
#define LP_OFF    (BZ_OFF + BZ_FLOATS)           // 3328
#define LP_FLOATS (LP_STEPS * 32 * 2)            // 1024
#define W1_OFF    (LP_OFF + LP_FLOATS)           // 4352
#define W1_FLOATS (W1_STEPS * 4 * 32 * 2)        // 2304
#define WPACK_FLOATS (W1_OFF + W1_FLOATS)        // 6656  (26624 bytes)

// ---------------------------------------------------------------------------
// Prologue: swizzle weights into per-(step,lane) WMMA-B operand pairs.
// Lane L (half = L>>4, n = L&15) at K-step t needs B[k][n] for k = 4t+2*half
// and k+1, with out-of-range columns/rows zeroed.
// ---------------------------------------------------------------------------
__global__ void pnn_pack_weights(const float* __restrict__ lzw,
                                 const float* __restrict__ lpw,
                                 const float* __restrict__ W1,
                                 float* __restrict__ ws)
{
    const int stride = gridDim.x * blockDim.x;
    const int tid = blockIdx.x * blockDim.x + threadIdx.x;

    for (int e = tid; e < BZ_STEPS * 32; e += stride) {        // lz weights
        int t = e >> 5, L = e & 31;
        int half = L >> 4, d = L & 15;
        int ks = 4 * t + 2 * half;
        float v0 = (d < DPROD) ? lzw[d * K_EMB + ks]     : 0.0f;
        float v1 = (d < DPROD) ? lzw[d * K_EMB + ks + 1] : 0.0f;
        ws[BZ_OFF + 2 * e]     = v0;
        ws[BZ_OFF + 2 * e + 1] = v1;
    }
    for (int e = tid; e < LP_STEPS * 32; e += stride) {        // lp weights
        int t = e >> 5, L = e & 31;
        int half = L >> 4, d = L & 15;
        int ks = 4 * t + 2 * half;
        float v0 = (d < DPROD) ? lpw[d * (EMBED * EMBED) + ks]     : 0.0f;
        float v1 = (d < DPROD) ? lpw[d * (EMBED * EMBED) + ks + 1] : 0.0f;
        ws[LP_OFF + 2 * e]     = v0;
        ws[LP_OFF + 2 * e + 1] = v1;
    }
    for (int e = tid; e < W1_STEPS * 4 * 32; e += stride) {    // W1 (4 N-tiles)
        int t = e >> 7, rem = e & 127;
        int nt = rem >> 5, L = rem & 31;
        int half = L >> 4, n = nt * 16 + (L & 15);
        int ks = 4 * t + 2 * half;
        float v0 = (ks     < H_DIM) ? W1[ks       * HIDDEN + n] : 0.0f;
        float v1 = (ks + 1 < H_DIM) ? W1[(ks + 1) * HIDDEN + n] : 0.0f;
        ws[W1_OFF + 2 * e]     = v0;
        ws[W1_OFF + 2 * e + 1] = v1;
    }
}

// ---------------------------------------------------------------------------
// Main fused kernel.
// ---------------------------------------------------------------------------
__global__ __launch_bounds__(32 * WAVES_PER_BLOCK)
void pnn_fused_wmma(const int* __restrict__ x,
                    const float* __restrict__ emb_tables,
                    const float* __restrict__ wpack,  // packed weights in d_ws
                    const float* __restrict__ b1,     // (64)
                    const float* __restrict__ W2,     // (64, 1)
                    const float* __restrict__ b2,     // (1)
                    float* __restrict__ out,          // (B)
                    int B, int vocab)
{
    __shared__ float w_lds[WPACK_FLOATS];                    // 26624 B, per block
    __shared__ float lds_emb[WAVES_PER_BLOCK][TILE][K_EMB];  // gathered embeddings
    __shared__ float lds_s  [WAVES_PER_BLOCK][TILE][EMBED];  // s = sum_n emb
    __shared__ float lds_h  [WAVES_PER_BLOCK][TILE][H_PAD];  // [lz | lp | dense | 0]

    const int lane = threadIdx.x & 31;
    const int wave = threadIdx.x >> 5;
    const int tile = blockIdx.x * WAVES_PER_BLOCK + wave;
    const int row0 = tile * TILE;

    float (*emb)[K_EMB] = lds_emb[wave];
    float (*sbuf)[EMBED] = lds_s[wave];
    float (*hbuf)[H_PAD] = lds_h[wave];

    // prefetch this tile's index rows (global_prefetch_b8) while staging weights
    {
        int pr = row0 + (lane & (TILE - 1));
        if (pr >= B) pr = B - 1;
        __builtin_prefetch(x + (size_t)pr * NFEAT, 0, 1);
    }

    // stage packed weights into LDS (whole block, float4 copies)
    for (int i = threadIdx.x; i < WPACK_FLOATS / 4; i += 32 * WAVES_PER_BLOCK) {
        ((float4*)w_lds)[i] = ((const float4*)wpack)[i];
    }

    // zero the K padding columns (33..35) of h
    for (int i = lane; i < TILE * (H_PAD - H_DIM); i += 32) {
        int r = i / (H_PAD - H_DIM);
        int c = H_DIM + i % (H_PAD - H_DIM);
        hbuf[r][c] = 0.0f;
    }

    // ---- Phase 1: embedding gather (16 rows x 26 tables, 13 gathers/lane) ----
    #pragma unroll
    for (int it = 0; it < (TILE * N_EMB) / 32; ++it) {       // 13 iterations
        int p = it * 32 + lane;
        int r = p / N_EMB;
        int n = p - r * N_EMB;
        int brow = row0 + r; if (brow >= B) brow = B - 1;
        int cat = x[brow * NFEAT + n];
        const float* src = emb_tables + ((size_t)n * (size_t)vocab + (size_t)cat) * EMBED;
        float4 lo = *(const float4*)(src);
        float4 hi = *(const float4*)(src + 4);
        *(float4*)(&emb[r][n * EMBED])     = lo;
        *(float4*)(&emb[r][n * EMBED + 4]) = hi;
    }
    // dense features -> h columns 20..32
    for (int i = lane; i < TILE * N_DENSE; i += 32) {
        int r = i / N_DENSE, j = i - r * N_DENSE;
        int brow = row0 + r; if (brow >= B) brow = B - 1;
        hbuf[r][2 * DPROD + j] = (float)x[brow * NFEAT + N_EMB + j];
    }
    __syncthreads();

    // ---- Phase 2: s[r][m] = sum_n emb[r][n*8+m]  (float4-vectorized) ----
    {
        int r  = lane >> 1;            // 16 rows
        int m4 = (lane & 1) * 4;       // 2 float4 halves
        float4 acc = make_float4(0.f, 0.f, 0.f, 0.f);
        #pragma unroll
        for (int n = 0; n < N_EMB; ++n) {
            float4 v = *(const float4*)(&emb[r][n * EMBED + m4]);
            acc.x += v.x; acc.y += v.y; acc.z += v.z; acc.w += v.w;
        }
        *(float4*)(&sbuf[r][m4]) = acc;
    }
    __syncthreads();

    const int half = lane >> 4;     // 0: K sub 0/1, C rows 0..7 | 1: K sub 2/3, C rows 8..15
    const int l16  = lane & 15;     // M (A) / N (B,C)

    // ---- Phase 3: lz(16x10) = emb(16x208) @ LZ^T, N padded to 16 ----
    v8f accz = {};
    const float* bz = w_lds + BZ_OFF + 2 * lane;
    #pragma unroll 4
    for (int t = 0; t < BZ_STEPS; ++t) {                     // 52 WMMA steps
        int ks = 4 * t + 2 * half;
        v2f a  = *(const v2f*)(&emb[l16][ks]);
        v2f bm = *(const v2f*)(bz + t * 64);
        accz = __builtin_amdgcn_wmma_f32_16x16x4_f32(false, a, false, bm,
                                                     (short)0, accz, false, false);
    }
    if (l16 < DPROD) {
        #pragma unroll
        for (int v = 0; v < 8; ++v) hbuf[v + 8 * half][l16] = accz[v];
    }

    // ---- Phase 4: lp(16x10) = (s⊗s)(16x64) @ LP^T, N padded to 16 ----
    // Lane's s-row in registers; per-step A built with 2 mults (no LDS in loop).
    float sA[8];
    {
        float4 lo = *(const float4*)(&sbuf[l16][0]);
        float4 hi = *(const float4*)(&sbuf[l16][4]);
        sA[0] = lo.x; sA[1] = lo.y; sA[2] = lo.z; sA[3] = lo.w;
        sA[4] = hi.x; sA[5] = hi.y; sA[6] = hi.z; sA[7] = hi.w;
    }
    // k-step t: hi index = t>>1 (compile-time); lo index = (4t&7) + 2*half
    const float pe0 = half ? sA[2] : sA[0];   // t even: k&7 = 2*half, 2*half+1
    const float pe1 = half ? sA[3] : sA[1];
    const float po0 = half ? sA[6] : sA[4];   // t odd:  k&7 = 4+2*half, ...
    const float po1 = half ? sA[7] : sA[5];
    v8f accp = {};
    const float* bp = w_lds + LP_OFF + 2 * lane;
    #pragma unroll
    for (int t = 0; t < LP_STEPS; ++t) {                     // 16 WMMA steps
        float shi = sA[t >> 1];
        v2f a;
        a.x = shi * ((t & 1) ? po0 : pe0);
        a.y = shi * ((t & 1) ? po1 : pe1);
        v2f bm = *(const v2f*)(bp + t * 64);
        accp = __builtin_amdgcn_wmma_f32_16x16x4_f32(false, a, false, bm,
                                                     (short)0, accp, false, false);
    }
    if (l16 < DPROD) {
        #pragma unroll
        for (int v = 0; v < 8; ++v) hbuf[v + 8 * half][DPROD + l16] = accp[v];
    }
    __syncthreads();

    // ---- Phase 5: z(16x64) = h(16x36, zero-padded) @ W1(36x64) ----
    v8f acc[4] = {};                                         // 4 N-tiles of 16
    const float* bw = w_lds + W1_OFF + 2 * lane;
    #pragma unroll
    for (int t = 0; t < W1_STEPS; ++t) {                     // 9 K-steps x 4 tiles
        int ks = 4 * t + 2 * half;
        v2f a = *(const v2f*)(&hbuf[l16][ks]);
        #pragma unroll
        for (int nt = 0; nt < 4; ++nt) {
            v2f bm = *(const v2f*)(bw + (t * 4 + nt) * 64);
            acc[nt] = __builtin_amdgcn_wmma_f32_16x16x4_f32(false, a, false, bm,
                                                            (short)0, acc[nt], false, false);
        }
    }

    // ---- Phase 6: bias + relu, layer 2 (dot with W2), lane reduce, sigmoid ----
    float part[8];
    #pragma unroll
    for (int v = 0; v < 8; ++v) part[v] = 0.0f;
    #pragma unroll
    for (int nt = 0; nt < 4; ++nt) {
        int n = nt * 16 + l16;
        float bias = b1[n];
        float w2   = W2[n];
        #pragma unroll
        for (int v = 0; v < 8; ++v) {
            float z = acc[nt][v] + bias;
            z = z > 0.0f ? z : 0.0f;
            part[v] += z * w2;
        }
    }
    // reduce over the 16 lanes of each half (rows v + 8*half)
    #pragma unroll
    for (int m = 1; m < 16; m <<= 1) {
        #pragma unroll
        for (int v = 0; v < 8; ++v)
            part[v] += __shfl_xor(part[v], m, 32);
    }
    if (l16 == 0) {
        float bb = b2[0];
        #pragma unroll
        for (int v = 0; v < 8; ++v) {
            int r = row0 + v + 8 * half;
            if (r < B) {
                float zv = part[v] + bb;
                out[r] = 1.0f / (1.0f + __expf(-zv));
            }
        }
    }
}

extern "C" void kernel_launch(void* const* d_in, const int* in_sizes, int n_in,
                              void* d_out, int out_size, void* d_ws, size_t ws_size,
                              hipStream_t stream) {
    const int*   x   = (const int*)d_in[0];
    const float* emb = (const float*)d_in[1];
    const float* lzw = (const float*)d_in[2];
    const float* lpw = (const float*)d_in[3];
    const float* W1  = (const float*)d_in[4];
    const float* b1  = (const float*)d_in[5];
    const float* W2  = (const float*)d_in[6];
    const float* b2  = (const float*)d_in[7];
    float* wsf = (float*)d_ws;

    const int B     = in_sizes[0] / NFEAT;
    const int vocab = (int)((size_t)in_sizes[1] / (N_EMB * EMBED));

    // 1) swizzle weights into WMMA-B operand layout in d_ws
    pnn_pack_weights<<<8, 256, 0, stream>>>(lzw, lpw, W1, wsf);

    // 2) fused forward
    const int tiles  = (B + TILE - 1) / TILE;
    const int blocks = (tiles + WAVES_PER_BLOCK - 1) / WAVES_PER_BLOCK;
    pnn_fused_wmma<<<blocks, 32 * WAVES_PER_BLOCK, 0, stream>>>(
        x, emb, wsf, b1, W2, b2, (float*)d_out, B, vocab);
}